// deforming_medium_60473139528284
// MI455X (gfx1250) — compile-verified
//
#include <hip/hip_runtime.h>
#include <hip/hip_bf16.h>

// ---------------------------------------------------------------- constants
#define Wd   512
#define Hd   512
#define Bn   64
#define HWd  (Hd * Wd)
#define SCALE_XY (4.0f / 511.0f)        // IDENTITY_OFFSET/(W-1), H==W
#define DEFORM_OFF (1.0f + 2.0f/511.0f) // (2 + 4/511)/2
#define PBOUND 0.03125f                 // (2/W)*EPS = 2/512*8

typedef __attribute__((ext_vector_type(2))) float v2f;
typedef __attribute__((ext_vector_type(8))) float v8f;
typedef __attribute__((ext_vector_type(4))) unsigned int u32x4;
typedef __attribute__((ext_vector_type(8))) int i32x8;
typedef __attribute__((ext_vector_type(4))) int i32x4;

// Normalized 13-tap Gaussian (sigma = 13*0.15+0.35 = 2.3), zero-padded so
// the Toeplitz lookup T[k][n] = WPAD[k-n+15] is branch-free for
// k-n in [-15, 27].
__constant__ float WPAD[43] = {
    0.f, 0.f, 0.f, 0.f, 0.f, 0.f, 0.f, 0.f, 0.f, 0.f, 0.f, 0.f, 0.f, 0.f, 0.f,
    0.0057988f, 0.0164012f, 0.0383987f, 0.0744143f, 0.1193712f, 0.1585060f,
    0.1742186f,
    0.1585060f, 0.1193712f, 0.0744143f, 0.0383987f, 0.0164012f, 0.0057988f,
    0.f, 0.f, 0.f, 0.f, 0.f, 0.f, 0.f, 0.f, 0.f, 0.f, 0.f, 0.f, 0.f, 0.f, 0.f
};

__device__ __forceinline__ int reflect512(int i) {
    if (i < 0) i = -i;
    if (i >= Wd) i = 2 * Wd - 2 - i;    // 'reflect' (no edge repeat)
    return i;
}
// Toeplitz band, branch-free: T[k][n] = w[k-n] (zero outside the band)
__device__ __forceinline__ float toep(int k, int n) {
    return WPAD[k - n + 15];
}

// ------------------------------------------------- horizontal blur via WMMA
// out[r,c] = sum_j in[r, col0-6+j] * T[j,c] : (16x28)*(28x16), 7x K=4 WMMA
__global__ __launch_bounds__(256) void hblur_wmma(const float* __restrict__ src,
                                                  float* __restrict__ dst) {
    __shared__ float lds[8][16 * 28];
    const int lane = threadIdx.x & 31;
    const int wave = threadIdx.x >> 5;
    const int t = blockIdx.x * 8 + wave;       // tile id
    const int img = t >> 10;                   // 1024 tiles per 512x512 image
    const int tile = t & 1023;
    const int row0 = (tile >> 5) << 4;
    const int col0 = (tile & 31) << 4;
    const float* base = src + (size_t)img * HWd;
    float* obase = dst + (size_t)img * HWd;

    for (int idx = lane; idx < 16 * 28; idx += 32) {
        int r = idx / 28, c = idx - r * 28;
        lds[wave][idx] = base[(row0 + r) * Wd + reflect512(col0 - 6 + c)];
    }
    __syncthreads();

    const int mn = lane & 15;                  // M for A lanes, N for B lanes
    const int kofs = (lane < 16) ? 0 : 2;      // K pair selector per half-wave
    v8f acc = {};
#pragma unroll
    for (int j = 0; j < 7; ++j) {
        const int k0 = 4 * j + kofs;
        v2f a, b;
        a.x = lds[wave][mn * 28 + k0];
        a.y = lds[wave][mn * 28 + k0 + 1];
        b.x = toep(k0, mn);
        b.y = toep(k0 + 1, mn);
        acc = __builtin_amdgcn_wmma_f32_16x16x4_f32(false, a, false, b,
                                                    (short)0, acc, false, false);
    }
    const int orow = row0 + ((lane < 16) ? 0 : 8);
    const int ocol = col0 + mn;
#pragma unroll
    for (int i = 0; i < 8; ++i)
        obase[(orow + i) * Wd + ocol] = acc[i];
}

// --------------------------------------------------- vertical blur via WMMA
// out[r,c] = sum_j T[j,r] * in[row0-6+j, c] : A = Toeplitz^T, B = input slab
__global__ __launch_bounds__(256) void vblur_wmma(const float* __restrict__ src,
                                                  float* __restrict__ dst) {
    __shared__ float lds[8][28 * 16];
    const int lane = threadIdx.x & 31;
    const int wave = threadIdx.x >> 5;
    const int t = blockIdx.x * 8 + wave;
    const int img = t >> 10;
    const int tile = t & 1023;
    const int row0 = (tile >> 5) << 4;
    const int col0 = (tile & 31) << 4;
    const float* base = src + (size_t)img * HWd;
    float* obase = dst + (size_t)img * HWd;

    for (int idx = lane; idx < 28 * 16; idx += 32) {
        int r = idx >> 4, c = idx & 15;
        lds[wave][idx] = base[reflect512(row0 - 6 + r) * Wd + col0 + c];
    }
    __syncthreads();

    const int mn = lane & 15;
    const int kofs = (lane < 16) ? 0 : 2;
    v8f acc = {};
#pragma unroll
    for (int j = 0; j < 7; ++j) {
        const int k0 = 4 * j + kofs;
        v2f a, b;
        a.x = toep(k0, mn);                    // A[m][k] = w[k-m]
        a.y = toep(k0 + 1, mn);
        b.x = lds[wave][k0 * 16 + mn];
        b.y = lds[wave][(k0 + 1) * 16 + mn];
        acc = __builtin_amdgcn_wmma_f32_16x16x4_f32(false, a, false, b,
                                                    (short)0, acc, false, false);
    }
    const int orow = row0 + ((lane < 16) ? 0 : 8);
    const int ocol = col0 + mn;
#pragma unroll
    for (int i = 0; i < 8; ++i)
        obase[(orow + i) * Wd + ocol] = acc[i];
}

// ------------------------- diff -> relu -> cumsum -> clips (per row/column)
// One block per 512-element line. The line (row for ch0, stride-512 column
// for ch1) is fetched with a single Tensor Data Mover 2D-tile descriptor
// into LDS (tensor_load_to_lds, TENSORcnt), then scanned in LDS.
__global__ __launch_bounds__(512) void scan512(const float* __restrict__ blur,
                                               float* __restrict__ samp) {
    __shared__ float tdbuf[512];      // TDM destination
    __shared__ float sc[512];         // scan buffer
    const int t = threadIdx.x;        // position along the scan dimension
    const int line = blockIdx.x;      // fixed row (ch0) / column (ch1)
    const int b = blockIdx.y;
    const int ch = blockIdx.z;
    const float* src = blur + (size_t)(b * 2 + ch) * HWd;
    float* dst = samp + (size_t)(b * 2 + ch) * HWd;

    // ---- Tensor DMA descriptor (D#): 2D tile of the 512x512 f32 plane ----
    const size_t gaddr = (size_t)(uintptr_t)src +
                         ((ch == 0) ? (size_t)line * Wd * 4u : (size_t)line * 4u);
    const unsigned ldsoff = (unsigned)(uintptr_t)(void*)tdbuf; // LDS byte addr
    const unsigned tile0 = (ch == 0) ? 512u : 1u;   // contiguous extent
    const unsigned tile1 = (ch == 0) ? 1u : 512u;   // strided extent
    u32x4 g0;
    g0[0] = 1u;                                      // count=1, user load
    g0[1] = ldsoff;                                  // lds_addr (bytes)
    g0[2] = (unsigned)(gaddr & 0xffffffffu);         // global_addr[31:0]
    g0[3] = (unsigned)((gaddr >> 32) & 0x1ffffffu)   // global_addr[56:32]
            | (2u << 30);                            // type=2 ("image")
    i32x8 g1;
    g1[0] = (int)(2u << 16);          // workgroup_mask=0, data_size=4B
    g1[1] = (int)(512u << 16);        // tensor_dim0[15:0]=512
    g1[2] = (int)(512u << 16);        // tensor_dim0[31:16]=0, tensor_dim1[15:0]=512
    g1[3] = (int)(tile0 << 16);       // tensor_dim1[31:16]=0, tile_dim0
    g1[4] = (int)tile1;               // tile_dim1, tile_dim2=0
    g1[5] = 512;                      // tensor_dim0_stride[31:0]=512 elements
    g1[6] = 0;                        // stride0[47:32]=0, stride1[15:0]=0
    g1[7] = 0;
    const i32x4 z4 = {0, 0, 0, 0};    // 2D tensor: groups 2/3 unused
    const i32x8 z8 = {0, 0, 0, 0, 0, 0, 0, 0};
    if (threadIdx.x < 32) {           // TDM ignores EXEC; issue from one wave
        __builtin_amdgcn_tensor_load_to_lds(g0, g1, z4, z4, z8, 0);
        __builtin_amdgcn_s_wait_tensorcnt(0);
    }
    __syncthreads();

    // ---- diff / relu / inclusive scan / clips ----
    const float iden  = (2.0f * (float)t - 511.0f) * (1.0f / 511.0f);
    const float sval  = tdbuf[t] + iden + DEFORM_OFF;
    float prev = 0.0f;
    if (t > 0) {
        const float idp = (2.0f * (float)(t - 1) - 511.0f) * (1.0f / 511.0f);
        prev = tdbuf[t - 1] + idp + DEFORM_OFF;
    }
    float a = fmaxf((sval - prev) * (1.0f / SCALE_XY), 0.0f); // relu(diff/scale)
    sc[t] = a;
    __syncthreads();
#pragma unroll
    for (int o = 1; o < 512; o <<= 1) {       // Hillis–Steele inclusive scan
        float v = sc[t];
        if (t >= o) v += sc[t - o];
        __syncthreads();
        sc[t] = v;
        __syncthreads();
    }
    float p = sc[t] * SCALE_XY - DEFORM_OFF - iden;           // accu2samp - iden
    p = fminf(fmaxf(p, -PBOUND), PBOUND);                     // prim_clip
    const size_t off = (ch == 0) ? ((size_t)line * Wd + t)
                                 : ((size_t)t * Wd + line);
    dst[off] = fminf(fmaxf(p + iden, -1.0f), 1.0f);           // samp_clip
}

// --------------------------------------------- bilinear grid sample (zeros)
__global__ __launch_bounds__(256) void gsample(const float* __restrict__ img,
                                               const float* __restrict__ samp,
                                               float* __restrict__ out) {
    const size_t idx = (size_t)blockIdx.x * 256 + threadIdx.x;  // B*H*W threads
    const int x = (int)(idx & 511);
    const int y = (int)((idx >> 9) & 511);
    const int b = (int)(idx >> 18);
    const float* sb = samp + (size_t)b * 2 * HWd;
    const float gx = sb[y * Wd + x];
    const float gy = sb[HWd + y * Wd + x];

    const float fx = (gx + 1.0f) * 0.5f * 511.0f;
    const float fy = (gy + 1.0f) * 0.5f * 511.0f;
    const float x0f = floorf(fx), y0f = floorf(fy);
    const float wx = fx - x0f, wy = fy - y0f;
    const int x0 = (int)x0f, y0 = (int)y0f;
    const int x1 = x0 + 1, y1 = y0 + 1;

    const float vx0 = (x0 >= 0 && x0 < Wd) ? 1.f : 0.f;
    const float vx1 = (x1 >= 0 && x1 < Wd) ? 1.f : 0.f;
    const float vy0 = (y0 >= 0 && y0 < Hd) ? 1.f : 0.f;
    const float vy1 = (y1 >= 0 && y1 < Hd) ? 1.f : 0.f;
    const int x0c = min(max(x0, 0), Wd - 1), x1c = min(max(x1, 0), Wd - 1);
    const int y0c = min(max(y0, 0), Hd - 1), y1c = min(max(y1, 0), Hd - 1);

    const float w00 = (1.f - wx) * (1.f - wy) * vx0 * vy0;
    const float w01 = wx * (1.f - wy) * vx1 * vy0;
    const float w10 = (1.f - wx) * wy * vx0 * vy1;
    const float w11 = wx * wy * vx1 * vy1;

    const float* ib = img + (size_t)b * 3 * HWd;
    float* ob = out + (size_t)b * 3 * HWd;
#pragma unroll
    for (int c = 0; c < 3; ++c) {
        const float* ic = ib + (size_t)c * HWd;
        ob[(size_t)c * HWd + y * Wd + x] =
            ic[y0c * Wd + x0c] * w00 + ic[y0c * Wd + x1c] * w01 +
            ic[y1c * Wd + x0c] * w10 + ic[y1c * Wd + x1c] * w11;
    }
}

// ------------------------------------------------------------------- launch
extern "C" void kernel_launch(void* const* d_in, const int* in_sizes, int n_in,
                              void* d_out, int out_size, void* d_ws, size_t ws_size,
                              hipStream_t stream) {
    const float* image = (const float*)d_in[0];   // [64,3,512,512] f32
    const float* prim  = (const float*)d_in[1];   // [64,2,512,512] f32
    float* out = (float*)d_out;                   // [64,3,512,512] f32
    float* ws0 = (float*)d_ws;                    // [64,2,512,512] f32 scratch
    float* tmpb = out;                            // blur intermediate reuses d_out
                                                  // (B*2*HW < out_size); final
                                                  // kernel overwrites it fully.

    const int blur_blocks = (Bn * 2 * 1024) / 8;  // wave-per-16x16-tile, 8 waves/block
    hblur_wmma<<<blur_blocks, 256, 0, stream>>>(prim, ws0);
    vblur_wmma<<<blur_blocks, 256, 0, stream>>>(ws0, tmpb);

    dim3 sg(512, Bn, 2);                          // line x batch x channel
    scan512<<<sg, 512, 0, stream>>>(tmpb, ws0);

    gsample<<<(Bn * HWd) / 256, 256, 0, stream>>>(image, ws0, out);
}